// DecoderRNN_79018808312087
// MI455X (gfx1250) — compile-verified
//
#include <hip/hip_runtime.h>
#include <hip/hip_bf16.h>
#include <math.h>

#define NHID 1024
#define NOUT 50257
#define KDIM 1024           // reduction dim, fixed = NHID
#define KTILE 128           // K-chunk staged in LDS per wave
#define TS 132              // LDS row stride (floats): 16B aligned, bank-conflict-free
#define WAVES 4             // waves per block in gemv
#define ROWS_PER_BLOCK (WAVES * 16)

typedef __attribute__((ext_vector_type(2))) float v2f;
typedef __attribute__((ext_vector_type(8))) float v8f;

// ---------------------------------------------------------------------------
// y[r] = dot(M[r, :], v) + bias[r]   for r in [0, nrows)
// Each wave computes 16 rows using V_WMMA_F32_16X16X4_F32 (full f32 precision).
// Matrix tiles (16 x 128) are staged LDS-coalesced; vector is block-shared.
// ---------------------------------------------------------------------------
__global__ void __launch_bounds__(WAVES * 32)
gemv_wmma_kernel(const float* __restrict__ M, const float* __restrict__ v,
                 const float* __restrict__ bias, float* __restrict__ y,
                 int nrows) {
    __shared__ float vlds[KDIM];
    __shared__ float tiles[WAVES][16 * TS];

    const int tid  = threadIdx.x;   // 0..127
    const int wave = tid >> 5;
    const int lane = tid & 31;

    // Load the vector once per block (coalesced float4).
    for (int i = tid; i < KDIM / 4; i += blockDim.x)
        ((float4*)vlds)[i] = ((const float4*)v)[i];
    __syncthreads();

    float* tile = tiles[wave];
    const int row0 = blockIdx.x * ROWS_PER_BLOCK + wave * 16;

    const int m    = lane & 15;           // A-fragment row
    const int koff = (lane >> 4) << 1;    // 0 for lanes 0-15, 2 for lanes 16-31

    v8f c = {};
    for (int k0 = 0; k0 < KDIM; k0 += KTILE) {
        // Stage 16 x 128 tile: iteration i = row, lane = float4 column.
        for (int i = 0; i < 16; ++i) {
            int rg = row0 + i;
            if (rg > nrows - 1) rg = nrows - 1;           // clamp (dup reads OK)
            float4 val = *(const float4*)(M + (size_t)rg * KDIM + k0 + lane * 4);
            *(float4*)(tile + i * TS + lane * 4) = val;
        }
        __syncthreads();   // uniform trip count across all waves

        for (int k2 = 0; k2 < KTILE; k2 += 4) {
            // A: 16x4 f32 fragment per documented layout
            float2 af = *(const float2*)(tile + m * TS + k2 + koff);
            // B: 4x16 with x[k] broadcast into all 16 columns
            float2 bf = *(const float2*)(vlds + k0 + k2 + koff);
            v2f a; a.x = af.x; a.y = af.y;
            v2f b; b.x = bf.x; b.y = bf.y;
            c = __builtin_amdgcn_wmma_f32_16x16x4_f32(
                    false, a, false, b, (short)0, c, false, false);
        }
        __syncthreads();
    }

    // C/D layout: VGPR j -> M=j (lanes 0-15), M=8+j (lanes 16-31); all N equal.
    if (lane == 0 || lane == 16) {
        int base = row0 + ((lane >> 4) << 3);
        for (int j = 0; j < 8; ++j) {
            int r = base + j;
            if (r < nrows) y[r] = c[j] + bias[r];
        }
    }
}

// ---------------------------------------------------------------------------
__global__ void embed_relu_kernel(const int* __restrict__ token,
                                  const float* __restrict__ emb,
                                  const float* __restrict__ hidden,
                                  float* __restrict__ x, float* __restrict__ hbuf) {
    int i = blockIdx.x * blockDim.x + threadIdx.x;
    if (i < NHID) {
        float e = emb[(size_t)token[0] * NHID + i];
        x[i]    = e > 0.0f ? e : 0.0f;
        hbuf[i] = hidden[i];
    }
}

__device__ __forceinline__ float sigmoidf_(float v) {
    return 1.0f / (1.0f + __expf(-v) * 0.0f + expf(-v) * 1.0f);  // use precise expf
}

__global__ void gru_gate_kernel(const float* __restrict__ gi,
                                const float* __restrict__ gh,
                                float* __restrict__ hbuf, float* __restrict__ x) {
    int i = blockIdx.x * blockDim.x + threadIdx.x;
    if (i < NHID) {
        float r = 1.0f / (1.0f + expf(-(gi[i] + gh[i])));
        float z = 1.0f / (1.0f + expf(-(gi[NHID + i] + gh[NHID + i])));
        float n = tanhf(gi[2 * NHID + i] + r * gh[2 * NHID + i]);
        float h = (1.0f - z) * n + z * hbuf[i];
        hbuf[i] = h;
        x[i]    = h;
    }
}

__global__ void __launch_bounds__(1024)
logsoftmax_kernel(const float* __restrict__ logits,
                  const float* __restrict__ hbuf, float* __restrict__ out) {
    __shared__ float red[1024];
    const int t = threadIdx.x;

    float mx = -INFINITY;
    for (int i = t; i < NOUT; i += 1024) mx = fmaxf(mx, logits[i]);
    red[t] = mx; __syncthreads();
    for (int s = 512; s > 0; s >>= 1) {
        if (t < s) red[t] = fmaxf(red[t], red[t + s]);
        __syncthreads();
    }
    mx = red[0]; __syncthreads();

    float sum = 0.0f;
    for (int i = t; i < NOUT; i += 1024) sum += expf(logits[i] - mx);
    red[t] = sum; __syncthreads();
    for (int s = 512; s > 0; s >>= 1) {
        if (t < s) red[t] += red[t + s];
        __syncthreads();
    }
    const float lse = mx + logf(red[0]);

    for (int i = t; i < NOUT; i += 1024) out[i] = logits[i] - lse;
    out[NOUT + t] = hbuf[t];   // second output: final hidden state
}

// ---------------------------------------------------------------------------
extern "C" void kernel_launch(void* const* d_in, const int* in_sizes, int n_in,
                              void* d_out, int out_size, void* d_ws, size_t ws_size,
                              hipStream_t stream) {
    const int*   token  = (const int*)  d_in[0];
    const float* hidden = (const float*)d_in[1];
    const float* emb    = (const float*)d_in[2];
    const float* w_ih   = (const float*)d_in[3];
    const float* w_hh   = (const float*)d_in[4];
    const float* b_ih   = (const float*)d_in[5];
    const float* b_hh   = (const float*)d_in[6];
    const float* W_out  = (const float*)d_in[7];
    const float* b_out  = (const float*)d_in[8];
    float* out = (float*)d_out;

    float* ws     = (float*)d_ws;
    float* x      = ws;                 // NHID
    float* hbuf   = ws + NHID;          // NHID
    float* gi     = ws + 2 * NHID;      // 3*NHID
    float* gh     = ws + 5 * NHID;      // 3*NHID
    float* logits = ws + 8 * NHID;      // NOUT

    embed_relu_kernel<<<(NHID + 255) / 256, 256, 0, stream>>>(token, emb, hidden, x, hbuf);

    const int nbGate = (3 * NHID + ROWS_PER_BLOCK - 1) / ROWS_PER_BLOCK;   // 48
    for (int layer = 0; layer < 2; ++layer) {
        gemv_wmma_kernel<<<nbGate, WAVES * 32, 0, stream>>>(w_ih, x,    b_ih, gi, 3 * NHID);
        gemv_wmma_kernel<<<nbGate, WAVES * 32, 0, stream>>>(w_hh, hbuf, b_hh, gh, 3 * NHID);
        gru_gate_kernel<<<(NHID + 255) / 256, 256, 0, stream>>>(gi, gh, hbuf, x);
    }

    const int nbOut = (NOUT + ROWS_PER_BLOCK - 1) / ROWS_PER_BLOCK;        // 786
    gemv_wmma_kernel<<<nbOut, WAVES * 32, 0, stream>>>(W_out, x, b_out, logits, NOUT);

    logsoftmax_kernel<<<1, 1024, 0, stream>>>(logits, hbuf, out);
}